// RecurrentFRAdamaxAttention_v2_18906446037670
// MI455X (gfx1250) — compile-verified
//
#include <hip/hip_runtime.h>

typedef __attribute__((ext_vector_type(2))) float v2f;
typedef __attribute__((ext_vector_type(8))) float v8f;

#define EPSC    1e-6f
#define STEPSZ  0.5f
#define BETAC   0.9f
#define DELTAC  0.001f
#define DDIM    64
#define MDIM    64
#define LDSTR   68   // row stride (floats) for Si tile in LDS: keeps float4 16B-aligned

__device__ __forceinline__ float feat(float x) {
    // elu(x) + 1
    return x > 0.0f ? x + 1.0f : __expf(x);
}

__global__ __launch_bounds__(256)
void fr_adamax_attn_kernel(const float* __restrict__ q_in, const float* __restrict__ k_in,
                           const float* __restrict__ v_in, const float* __restrict__ Sip,
                           const float* __restrict__ Zi,   const float* __restrict__ Pi,
                           const float* __restrict__ Mi,   const float* __restrict__ Uip,
                           const float* __restrict__ Sip2,
                           float* __restrict__ outV, float* __restrict__ outS,
                           float* __restrict__ outZ, float* __restrict__ outP,
                           float* __restrict__ outM, float* __restrict__ outU,
                           float* __restrict__ outSp)
{
    __shared__ __align__(16) float sq[DDIM];
    __shared__ __align__(16) float sk[DDIM];
    __shared__ __align__(16) float sv[MDIM];
    __shared__ __align__(16) float sSi[DDIM * LDSTR];
    __shared__ float sred[4];   // [0]=||Ui-Uiprev||^2  [1]=||Sip-Sip2||^2  [2]=Q.Zi_new

    const int tid = threadIdx.x;
    const int hid = blockIdx.x;                 // flat (n,h) head index
    const size_t vb = (size_t)hid * 64;         // vector base
    const size_t mb = (size_t)hid * 4096;       // matrix base

    // ---------------- phase 0: vectors ----------------
    if (tid < 4) sred[tid] = 0.0f;
    float qz = 0.0f;
    if (tid < DDIM) {
        float qv = feat(q_in[vb + tid]);
        float kv = feat(k_in[vb + tid]);
        float zn = Zi[vb + tid] + kv;
        sq[tid] = qv;
        sk[tid] = kv;
        outZ[vb + tid] = zn;
        qz = qv * zn;
    }
    if (tid >= 64 && tid < 128) sv[tid - 64] = v_in[vb + (tid - 64)];
    __syncthreads();
    if (tid < DDIM) atomicAdd(&sred[2], qz);

    // ---------------- phase A: norms (stream 3 matrices once) ----------------
    // thread layout: float4 index i4 = tid + 256*c  -> d = tid/16 + 16c, m = 4*(tid%16)
    const float4* pU  = (const float4*)(Uip  + mb);
    const float4* pS  = (const float4*)(Sip  + mb);
    const float4* pS2 = (const float4*)(Sip2 + mb);
    const float4* vv4 = (const float4*)sv;
    const int mcol = tid & 15;                  // float4 column in M

    float sipreg[16];   // Siprev kept in registers (read once)
    float uireg[16];    // Ui kept in registers (computed once)
    float accu = 0.0f, accs = 0.0f;

#pragma unroll
    for (int c = 0; c < 4; ++c) {
        const int i4 = tid + 256 * c;
        const int d  = (tid >> 4) + 16 * c;
        float4 u0 = pU[i4];
        float4 s0 = pS[i4];
        float4 s2 = pS2[i4];
        float  kd = sk[d];
        float4 vl = vv4[mcol];
        float ux = kd * vl.x, uy = kd * vl.y, uz = kd * vl.z, uw = kd * vl.w;
        uireg[4*c+0] = ux; uireg[4*c+1] = uy; uireg[4*c+2] = uz; uireg[4*c+3] = uw;
        sipreg[4*c+0] = s0.x; sipreg[4*c+1] = s0.y; sipreg[4*c+2] = s0.z; sipreg[4*c+3] = s0.w;
        float dx = ux - u0.x, dy = uy - u0.y, dz = uz - u0.z, dw = uw - u0.w;
        accu += dx*dx + dy*dy + dz*dz + dw*dw;
        float ex = s0.x - s2.x, ey = s0.y - s2.y, ez = s0.z - s2.z, ew = s0.w - s2.w;
        accs += ex*ex + ey*ey + ez*ez + ew*ew;
    }
    atomicAdd(&sred[0], accu);
    atomicAdd(&sred[1], accs);
    __syncthreads();

    // ---------------- mu (per-head scalar) ----------------
    float nu = sqrtf(sred[0]);
    float ns = sqrtf(sred[1]);
    float r  = sqrtf(STEPSZ * nu / ns);
    float t  = 1.0f - r;
    float mu = t * t;
    mu = fminf(fmaxf(mu, 0.0f), 1.0f - DELTAC);

    // ---------------- phase B: elementwise update (stream 2 in, 5 out) ----------------
    const float4* pP = (const float4*)(Pi + mb);
    const float4* pM = (const float4*)(Mi + mb);
    float4* oS  = (float4*)(outS  + mb);
    float4* oP  = (float4*)(outP  + mb);
    float4* oM  = (float4*)(outM  + mb);
    float4* oU  = (float4*)(outU  + mb);
    float4* oSp = (float4*)(outSp + mb);

#pragma unroll
    for (int c = 0; c < 4; ++c) {
        const int i4 = tid + 256 * c;
        const int d  = (tid >> 4) + 16 * c;
        float4 pi = pP[i4];
        float4 mi = pM[i4];
        float4 un = make_float4(uireg[4*c+0], uireg[4*c+1], uireg[4*c+2], uireg[4*c+3]);
        float4 sp = make_float4(sipreg[4*c+0], sipreg[4*c+1], sipreg[4*c+2], sipreg[4*c+3]);
        float4 pn, mn, si;
        pn.x = mu * pi.x - STEPSZ * un.x;  pn.y = mu * pi.y - STEPSZ * un.y;
        pn.z = mu * pi.z - STEPSZ * un.z;  pn.w = mu * pi.w - STEPSZ * un.w;
        mn.x = fmaxf(BETAC * mi.x, fabsf(un.x));  mn.y = fmaxf(BETAC * mi.y, fabsf(un.y));
        mn.z = fmaxf(BETAC * mi.z, fabsf(un.z));  mn.w = fmaxf(BETAC * mi.w, fabsf(un.w));
        si.x = sp.x - pn.x / sqrtf(mn.x + 1e-16f);
        si.y = sp.y - pn.y / sqrtf(mn.y + 1e-16f);
        si.z = sp.z - pn.z / sqrtf(mn.z + 1e-16f);
        si.w = sp.w - pn.w / sqrtf(mn.w + 1e-16f);
        oP[i4]  = pn;
        oM[i4]  = mn;
        oU[i4]  = un;
        oSp[i4] = sp;     // Siprev pass-through output (from registers, no re-read)
        oS[i4]  = si;
        *(float4*)&sSi[d * LDSTR + 4 * mcol] = si;   // stage Si for WMMA matvec
    }
    __syncthreads();

    // ---------------- phase C: V = Z * (Q^T Si) via V_WMMA_F32_16X16X4_F32 ----------------
    // A = Q replicated into all 16 rows -> every output row equals V segment.
    // f32 fragment layout (wave32): VGPR0 = {K0 | K2} across lane halves, VGPR1 = {K1 | K3};
    // B lane = output column; C/D VGPR0 = row0 (lanes 0-15) / row8 (lanes 16-31).
    if (tid < 32) {
        const float Zs  = 1.0f / (sred[2] + EPSC);
        const int half  = tid >> 4;    // 0: K+0/K+1, 1: K+2/K+3
        const int col   = tid & 15;
        v8f acc0 = {}, acc1 = {}, acc2 = {}, acc3 = {};
#pragma unroll
        for (int kc = 0; kc < 16; ++kc) {
            const int r0 = 4 * kc + 2 * half;
            v2f a, b0, b1, b2, b3;
            a[0]  = sq[r0];
            a[1]  = sq[r0 + 1];
            b0[0] = sSi[r0 * LDSTR +  0 + col];  b0[1] = sSi[(r0 + 1) * LDSTR +  0 + col];
            b1[0] = sSi[r0 * LDSTR + 16 + col];  b1[1] = sSi[(r0 + 1) * LDSTR + 16 + col];
            b2[0] = sSi[r0 * LDSTR + 32 + col];  b2[1] = sSi[(r0 + 1) * LDSTR + 32 + col];
            b3[0] = sSi[r0 * LDSTR + 48 + col];  b3[1] = sSi[(r0 + 1) * LDSTR + 48 + col];
            acc0 = __builtin_amdgcn_wmma_f32_16x16x4_f32(false, a, false, b0, (short)0, acc0, false, false);
            acc1 = __builtin_amdgcn_wmma_f32_16x16x4_f32(false, a, false, b1, (short)0, acc1, false, false);
            acc2 = __builtin_amdgcn_wmma_f32_16x16x4_f32(false, a, false, b2, (short)0, acc2, false, false);
            acc3 = __builtin_amdgcn_wmma_f32_16x16x4_f32(false, a, false, b3, (short)0, acc3, false, false);
        }
        if (half == 0) {
            outV[vb +  0 + col] = Zs * acc0[0];
            outV[vb + 16 + col] = Zs * acc1[0];
            outV[vb + 32 + col] = Zs * acc2[0];
            outV[vb + 48 + col] = Zs * acc3[0];
        }
    }
}

extern "C" void kernel_launch(void* const* d_in, const int* in_sizes, int n_in,
                              void* d_out, int out_size, void* d_ws, size_t ws_size,
                              hipStream_t stream) {
    const float* q_in = (const float*)d_in[0];
    const float* k_in = (const float*)d_in[1];
    const float* v_in = (const float*)d_in[2];
    const float* Sip  = (const float*)d_in[3];
    const float* Zi   = (const float*)d_in[4];
    const float* Pi   = (const float*)d_in[5];
    const float* Mi   = (const float*)d_in[6];
    const float* Uip  = (const float*)d_in[7];
    const float* Sip2 = (const float*)d_in[8];

    const size_t NH = (size_t)in_sizes[0] / 64;   // N*H heads (query is (N,H,64))

    float* out  = (float*)d_out;
    // return order: (V, Si, Zi_new, Pi_new, Mi_new, Ui, Siprev)
    float* outV  = out;
    float* outS  = outV + NH * 64;
    float* outZ  = outS + NH * 4096;
    float* outP  = outZ + NH * 64;
    float* outM  = outP + NH * 4096;
    float* outU  = outM + NH * 4096;
    float* outSp = outU + NH * 4096;

    fr_adamax_attn_kernel<<<dim3((unsigned)NH), dim3(256), 0, stream>>>(
        q_in, k_in, v_in, Sip, Zi, Pi, Mi, Uip, Sip2,
        outV, outS, outZ, outP, outM, outU, outSp);
}